// KAN_27101243638027
// MI455X (gfx1250) — compile-verified
//
#include <hip/hip_runtime.h>
#include <hip/hip_bf16.h>
#include <math.h>

// ---------------------------------------------------------------------------
// KAN collapsed forward:  out = sum_i Ws[i,-1] * ( silu(x_i) + sin(x_i) * sum_g Cs[i,-1,g] )
//
// Latency-bound scalar reduction over 4096 elements (~80 KB useful bytes).
// Single 1024-thread block (32 wave32's on one WGP) provides ample memory-
// level parallelism. Cross-lane reduction is done with V_WMMA_F32_16X16X4_F32
// against an all-ones B matrix: D row-sums == sum of the wave's partials,
// independent of the exact A-slot <-> lane mapping (each partial occupies
// exactly one A element, the rest are zero).
// ---------------------------------------------------------------------------

typedef __attribute__((ext_vector_type(2))) float v2f;
typedef __attribute__((ext_vector_type(8))) float v8f;

// Full wave32 sum of per-lane value p, returned in every lane.
// EXEC must be all 1s when this is called (we guarantee that structurally:
// every thread of the block executes it, no divergence around the call).
__device__ __forceinline__ float wave_sum32(float p) {
#if defined(__gfx1250__) && __has_builtin(__builtin_amdgcn_wmma_f32_16x16x4_f32)
    v2f a;
    a[0] = p;       // each lane contributes exactly one A element
    a[1] = 0.0f;    // remaining A slot zeroed
    v2f b;
    b[0] = 1.0f;    // B = all-ones 4x16  ->  D[m,n] = rowsum_m(A)
    b[1] = 1.0f;
    v8f c = {};
    // 8 args: (neg_a, A, neg_b, B, c_mod, C, reuse_a, reuse_b)
    c = __builtin_amdgcn_wmma_f32_16x16x4_f32(
        /*neg_a=*/false, a, /*neg_b=*/false, b,
        /*c_mod=*/(short)0, c, /*reuse_a=*/false, /*reuse_b=*/false);
    // Lanes 0-15 hold D rows 0..7 in c[0..7]; lanes 16-31 hold rows 8..15.
    float s = ((c[0] + c[1]) + (c[2] + c[3])) + ((c[4] + c[5]) + (c[6] + c[7]));
    // Combine the two half-wave partial row-sum groups (SWAPX16).
    s += __shfl_xor(s, 16, 32);
    return s;
#else
    for (int off = 16; off > 0; off >>= 1) p += __shfl_xor(p, off, 32);
    return p;
#endif
}

__global__ __launch_bounds__(1024) void kan_collapsed_sum(
    const float* __restrict__ x,
    const float* __restrict__ Ws,
    const float* __restrict__ Cs,
    float* __restrict__ out,
    int n, long long wcols, long long g) {
    __shared__ float warp_sums[32];

    const int tid  = (int)threadIdx.x;
    const int lane = tid & 31;
    const int wave = tid >> 5;

    float p = 0.0f;
    for (int i = tid; i < n; i += 1024) {
        const float xv = x[i];
        // last column of row i
        const long long rowoff = (long long)i * wcols + (wcols - 1);
        // streaming, single-use data: non-temporal loads
        const float w = __builtin_nontemporal_load(Ws + rowoff);
        const float* cp = Cs + rowoff * g;
        float cs = 0.0f;
        for (long long gg = 0; gg < g; ++gg)
            cs += __builtin_nontemporal_load(cp + gg);
        const float sig  = 1.0f / (1.0f + expf(-xv));   // sigmoid
        const float silu = xv * sig;
        p += w * (silu + sinf(xv) * cs);
    }

    // Stage 1: per-wave reduction (WMMA) -- all 32 waves, EXEC all-1s.
    const float wsum = wave_sum32(p);
    if (lane == 0) warp_sums[wave] = wsum;
    __syncthreads();

    // Stage 2: every wave redundantly reduces the 32 wave-sums (keeps EXEC
    // all-1s for the WMMA in every executing wave; no predication).
    const float total = wave_sum32(warp_sums[lane]);

    if (tid == 0) out[0] = total;
}

extern "C" void kernel_launch(void* const* d_in, const int* in_sizes, int n_in,
                              void* d_out, int out_size, void* d_ws, size_t ws_size,
                              hipStream_t stream) {
    const float* x  = (const float*)d_in[0];
    const float* Ws = (const float*)d_in[1];
    const float* Cs = (const float*)d_in[2];

    const int n = in_sizes[0];                                   // 4096
    const long long wcols = (long long)in_sizes[1] / n;          // 2N+1 = 8193
    const long long g     = (long long)in_sizes[2] / in_sizes[1];// 3

    kan_collapsed_sum<<<1, 1024, 0, stream>>>(x, Ws, Cs, (float*)d_out,
                                              n, wcols, g);
}